// CustomSTFT_12171937317491
// MI455X (gfx1250) — compile-verified
//
#include <hip/hip_runtime.h>
#include <math.h>

// ---------------------------------------------------------------------------
// STFT (N_FFT=4096, HOP=1024) via two-step Cooley-Tukey 64x64 DFT factorization
// mapped onto CDNA5 V_WMMA_F32_16X16X4_F32 (full fp32 matrix pipeline).
//   X[k1 + 64*k2] = sum_{n1} W4096^{n1*k1} * W64^{n1*k2} *
//                       [ sum_{n2} z[n1 + 64*n2] * W64^{n2*k1} ]
// Stage 1:  Y  = Xm * F64          (64x64 complex matmul, WMMA tiles)
// Twiddle:  Y' = Y .* W4096^{n1*k1}
// Stage 2:  Z  = F64 * Y'          (64x64 complex matmul, WMMA tiles)
// Output: out[0][b][f] = Re Z, out[1][b][f] = Im Z for b in [0, 2048].
// ---------------------------------------------------------------------------

#define N_FFT 4096
#define HOP   1024
#define NBINS 2049
#define PAD   (N_FFT / 2)

typedef __attribute__((ext_vector_type(2))) float v2f;
typedef __attribute__((ext_vector_type(8))) float v8f;

__device__ __forceinline__ v8f wmma4_f32(v2f a, v2f b, v8f c) {
  // 8 args: (neg_a, A, neg_b, B, c_mod, C, reuse_a, reuse_b)
  // f32 WMMA supports only C-negate per ISA; neg_a/neg_b must be false.
  return __builtin_amdgcn_wmma_f32_16x16x4_f32(
      false, a, false, b, (short)0, c, false, false);
}

__global__ __launch_bounds__(256)
void stft_wmma_kernel(const float* __restrict__ x,
                      const float* __restrict__ win,
                      float* __restrict__ out,
                      int T, int num_frames) {
  // Complex frame (stage 1 input), later reused as Y' (stage 2 input): 32 KB
  __shared__ float sZr[N_FFT];
  __shared__ float sZi[N_FFT];
  // 64-point DFT matrix F64 = cos - i*sin : 32 KB
  __shared__ float sFr[64 * 64];
  __shared__ float sFi[64 * 64];

  const int f    = blockIdx.x;
  const int tid  = threadIdx.x;
  const int lane = tid & 31;          // wave32
  const int wave = tid >> 5;          // 8 waves per block

  // ---- load frame with reflect padding, apply Hann window, pack complex ----
  const float* __restrict__ x0 = x;
  const float* __restrict__ x1 = x + T;
  #pragma unroll
  for (int i = 0; i < 16; ++i) {
    int j = tid + 256 * i;            // 0..4095
    int q = f * HOP + j - PAD;        // position in unpadded signal
    if (q < 0) q = -q;                // reflect left
    if (q >= T) q = 2 * T - 2 - q;    // reflect right
    float w = win[j];
    sZr[j] = x0[q] * w;
    sZi[j] = x1[q] * w;
  }

  // ---- precompute F64[r][c] = W64^{r*c} = cos - i*sin ----
  const float PI2_64 = 6.28318530717958647692f / 64.0f;
  #pragma unroll
  for (int i = 0; i < 16; ++i) {
    int idx = tid * 16 + i;           // 0..4095
    int r = idx >> 6, c = idx & 63;
    float s, cc;
    __sincosf(PI2_64 * (float)((r * c) & 63), &s, &cc);
    sFr[idx] = cc;
    sFi[idx] = -s;
  }
  __syncthreads();

  const int lm = lane & 15;           // M/N index within tile
  const int kb = (lane >> 4) * 2;     // K sub-offset for A/B fragments

  // =========================== Stage 1: Y = Xm * F64 =======================
  // Xm[r][c] = z[r + 64*c] ; tile (m0,n0) accumulated over K=64 in steps of 4.
  v8f acr[2], aci[2];
  #pragma unroll
  for (int tt = 0; tt < 2; ++tt) {
    const int tile = wave * 2 + tt;   // 16 tiles over 8 waves
    const int m0 = (tile >> 2) * 16;
    const int n0 = (tile & 3) * 16;
    v8f accr = {}; v8f acci = {};
    #pragma unroll 4
    for (int k0 = 0; k0 < 64; k0 += 4) {
      v2f ar, ai, nai, br, bi;
      const int am = m0 + lm;
      const int ak = k0 + kb;
      ar[0] = sZr[am + 64 * ak];        ar[1] = sZr[am + 64 * (ak + 1)];
      ai[0] = sZi[am + 64 * ak];        ai[1] = sZi[am + 64 * (ak + 1)];
      nai[0] = -ai[0];                  nai[1] = -ai[1];
      const int bk = k0 + kb;
      const int bn = n0 + lm;
      br[0] = sFr[bk * 64 + bn];        br[1] = sFr[(bk + 1) * 64 + bn];
      bi[0] = sFi[bk * 64 + bn];        bi[1] = sFi[(bk + 1) * 64 + bn];
      accr = wmma4_f32(ar,  br, accr);  // +Xr*Fr
      accr = wmma4_f32(nai, bi, accr);  // -Xi*Fi
      acci = wmma4_f32(ar,  bi, acci);  // +Xr*Fi
      acci = wmma4_f32(ai,  br, acci);  // +Xi*Fr
    }
    acr[tt] = accr; aci[tt] = acci;
  }

  // ---- twiddle + writeback Y' into reused z buffers (row-major 64x64) ----
  __syncthreads();                     // all reads of z complete
  const float PI2_N = 6.28318530717958647692f / (float)N_FFT;
  #pragma unroll
  for (int tt = 0; tt < 2; ++tt) {
    const int tile = wave * 2 + tt;
    const int m0 = (tile >> 2) * 16;
    const int n0 = (tile & 3) * 16;
    #pragma unroll
    for (int v = 0; v < 8; ++v) {
      const int r = m0 + v + ((lane >> 4) << 3);  // n1
      const int c = n0 + lm;                      // k1
      float s, cc;
      __sincosf(PI2_N * (float)((r * c) & (N_FFT - 1)), &s, &cc);
      const float yr = acr[tt][v], yi = aci[tt][v];
      sZr[r * 64 + c] = yr * cc + yi * s;   // Y' = Y * (cos - i*sin)
      sZi[r * 64 + c] = yi * cc - yr * s;
    }
  }
  __syncthreads();

  // =========================== Stage 2: Z = F64 * Y' =======================
  const long cstride = (long)NBINS * (long)num_frames;
  #pragma unroll
  for (int tt = 0; tt < 2; ++tt) {
    const int tile = wave * 2 + tt;
    const int m0 = (tile >> 2) * 16;   // k2 rows
    const int n0 = (tile & 3) * 16;    // k1 cols
    v8f accr = {}; v8f acci = {};
    #pragma unroll 4
    for (int k0 = 0; k0 < 64; k0 += 4) {
      v2f ar, ai, nai, br, bi;
      const int am = m0 + lm;
      const int ak = k0 + kb;
      ar[0] = sFr[am * 64 + ak];        ar[1] = sFr[am * 64 + ak + 1];
      ai[0] = sFi[am * 64 + ak];        ai[1] = sFi[am * 64 + ak + 1];
      nai[0] = -ai[0];                  nai[1] = -ai[1];
      const int bk = k0 + kb;
      const int bn = n0 + lm;
      br[0] = sZr[bk * 64 + bn];        br[1] = sZr[(bk + 1) * 64 + bn];
      bi[0] = sZi[bk * 64 + bn];        bi[1] = sZi[(bk + 1) * 64 + bn];
      accr = wmma4_f32(ar,  br, accr);  // +Fr*Y'r
      accr = wmma4_f32(nai, bi, accr);  // -Fi*Y'i
      acci = wmma4_f32(ar,  bi, acci);  // +Fr*Y'i
      acci = wmma4_f32(ai,  br, acci);  // +Fi*Y'r
    }
    // ---- scatter bins b = k1 + 64*k2 (keep b < 2049) ----
    #pragma unroll
    for (int v = 0; v < 8; ++v) {
      const int k2 = m0 + v + ((lane >> 4) << 3);
      const int k1 = n0 + lm;
      const int b  = k1 + 64 * k2;
      if (b < NBINS) {
        out[(long)b * num_frames + f]           = accr[v];
        out[cstride + (long)b * num_frames + f] = acci[v];
      }
    }
  }
}

extern "C" void kernel_launch(void* const* d_in, const int* in_sizes, int n_in,
                              void* d_out, int out_size, void* d_ws, size_t ws_size,
                              hipStream_t stream) {
  const float* x   = (const float*)d_in[0];   // (2, T) float32
  const float* win = (const float*)d_in[1];   // (4096,) float32
  float* out = (float*)d_out;                 // (2, 2049, num_frames) float32
  const int T = in_sizes[0] / 2;
  const int num_frames = T / HOP + 1;         // 4097
  stft_wmma_kernel<<<dim3(num_frames), dim3(256), 0, stream>>>(
      x, win, out, T, num_frames);
}